// BigAttn_60928406061566
// MI455X (gfx1250) — compile-verified
//
#include <hip/hip_runtime.h>

// ---------------------------------------------------------------------------
// Problem constants (match reference)
// ---------------------------------------------------------------------------
#define NNODE 20000
#define NEDGE 100000
#define NLAYER 25
#define NHEAD 2
#define NCH 32
#define NF 64          // NHEAD * NCH
#define DMLP 800       // NLAYER * NCH
#define EPSBN 1e-5f

#define MLP_WAVES 10
#define MLP_THREADS (MLP_WAVES * 32)   // 320
#define MLP_ROWS (MLP_WAVES * 16)      // 160 rows per block; 20000/160 = 125 blocks

typedef float v2f __attribute__((ext_vector_type(2)));
typedef float v8f __attribute__((ext_vector_type(8)));

// CAS-loop float atomic max (amax initialized to -3e38)
__device__ __forceinline__ void atomicMaxF(float* addr, float val) {
  int* ai = reinterpret_cast<int*>(addr);
  int old = *ai;
  while (__int_as_float(old) < val) {
    int assumed = old;
    old = atomicCAS(ai, assumed, __float_as_int(val));
    if (old == assumed) break;
  }
}

// ---------------------------------------------------------------------------
// K1: per-layer workspace init
// ---------------------------------------------------------------------------
__global__ __launch_bounds__(256) void k_init(float* __restrict__ aggr,
                                              float* __restrict__ cnt,
                                              float* __restrict__ lasum,
                                              float* __restrict__ amax,
                                              float* __restrict__ denom) {
  const int i = blockIdx.x * blockDim.x + threadIdx.x;
  if (i < NNODE * NF) aggr[i] = 0.0f;
  if (i < NNODE) { cnt[i] = 0.0f; lasum[i] = 0.0f; }
  if (i < NNODE * NHEAD) { denom[i] = 0.0f; amax[i] = -3.0e38f; }
}

// ---------------------------------------------------------------------------
// K2: node transforms xl = x@Wl + bl, xr = x@Wr + br   (K = 5, tiny)
// ---------------------------------------------------------------------------
__global__ __launch_bounds__(256) void k_transform(const float* __restrict__ x,
                                                   const float* __restrict__ Wl,
                                                   const float* __restrict__ bl,
                                                   const float* __restrict__ Wr,
                                                   const float* __restrict__ br,
                                                   float* __restrict__ xl,
                                                   float* __restrict__ xr, int l) {
  const int i = blockIdx.x * blockDim.x + threadIdx.x;
  if (i >= NNODE * NF) return;
  const int n = i >> 6, f = i & 63;
  const float* wl = Wl + l * 5 * NF + f;
  const float* wr = Wr + l * 5 * NF + f;
  float xv[5];
#pragma unroll
  for (int k = 0; k < 5; ++k) xv[k] = x[n * 5 + k];
  float sl = bl[l * NF + f], sr = br[l * NF + f];
#pragma unroll
  for (int k = 0; k < 5; ++k) {
    sl += xv[k] * wl[k * NF];
    sr += xv[k] * wr[k * NF];
  }
  xl[i] = sl;
  xr[i] = sr;
}

// ---------------------------------------------------------------------------
// K3: in-degree count + edge-weight sum per destination (for self-loop mean)
// ---------------------------------------------------------------------------
__global__ __launch_bounds__(256) void k_cnt(const int* __restrict__ ei,
                                             const float* __restrict__ ew,
                                             float* __restrict__ cnt,
                                             float* __restrict__ lasum, int l) {
  const int e = blockIdx.x * blockDim.x + threadIdx.x;
  if (e >= NEDGE) return;
  const int d = ei[l * 2 * NEDGE + NEDGE + e];
  atomicAdd(&cnt[d], 1.0f);
  atomicAdd(&lasum[d], ew[l * NEDGE + e]);
}

// ---------------------------------------------------------------------------
// K4: attention logits + segment max.  Edge id e in [0, E+N): e>=E => self loop
// ---------------------------------------------------------------------------
__global__ __launch_bounds__(256) void k_att1(const int* __restrict__ ei,
                                              const float* __restrict__ ew,
                                              const float* __restrict__ xl,
                                              const float* __restrict__ xr,
                                              const float* __restrict__ We,
                                              const float* __restrict__ att,
                                              const float* __restrict__ cnt,
                                              const float* __restrict__ lasum,
                                              float* __restrict__ alphabuf,
                                              float* __restrict__ amax, int l) {
  const int idx = blockIdx.x * blockDim.x + threadIdx.x;
  if (idx >= (NEDGE + NNODE) * NHEAD) return;
  const int h = idx & 1;
  const int e = idx >> 1;
  int s, d;
  float w;
  if (e < NEDGE) {
    s = ei[l * 2 * NEDGE + e];
    d = ei[l * 2 * NEDGE + NEDGE + e];
    w = ew[l * NEDGE + e];
  } else {
    s = d = e - NEDGE;
    w = lasum[s] / fmaxf(cnt[s], 1.0f);  // fill_value='mean'
  }
  const float* we = We + l * NF + h * NCH;
  const float* at = att + l * NF + h * NCH;
  const float* pl = xl + (size_t)s * NF + h * NCH;
  const float* pr = xr + (size_t)d * NF + h * NCH;
  float a = 0.0f;
#pragma unroll 8
  for (int c = 0; c < NCH; ++c) {
    float v = pl[c] + pr[c] + w * we[c];
    v = v > 0.0f ? v : 0.2f * v;   // leaky_relu(., 0.2)
    a += v * at[c];
  }
  alphabuf[idx] = a;
  atomicMaxF(&amax[d * NHEAD + h], a);
}

// ---------------------------------------------------------------------------
// K5: exp(alpha - amax[dst]) and segment-sum denominator
// ---------------------------------------------------------------------------
__global__ __launch_bounds__(256) void k_att2(const int* __restrict__ ei,
                                              float* __restrict__ alphabuf,
                                              const float* __restrict__ amax,
                                              float* __restrict__ denom, int l) {
  const int idx = blockIdx.x * blockDim.x + threadIdx.x;
  if (idx >= (NEDGE + NNODE) * NHEAD) return;
  const int h = idx & 1;
  const int e = idx >> 1;
  const int d = (e < NEDGE) ? ei[l * 2 * NEDGE + NEDGE + e] : (e - NEDGE);
  const float w = __expf(alphabuf[idx] - amax[d * NHEAD + h]);
  alphabuf[idx] = w;
  atomicAdd(&denom[d * NHEAD + h], w);
}

// ---------------------------------------------------------------------------
// K6: scatter messages:  aggr[d, f] += xl[s, f] * alpha_norm
// ---------------------------------------------------------------------------
__global__ __launch_bounds__(256) void k_scatter(const int* __restrict__ ei,
                                                 const float* __restrict__ xl,
                                                 const float* __restrict__ alphabuf,
                                                 const float* __restrict__ denom,
                                                 float* __restrict__ aggr, int l) {
  const int idx = blockIdx.x * blockDim.x + threadIdx.x;
  const int f = idx & 63;
  const int e = idx >> 6;
  if (e >= NEDGE + NNODE) return;
  const int h = f >> 5;
  int s, d;
  if (e < NEDGE) {
    s = ei[l * 2 * NEDGE + e];
    d = ei[l * 2 * NEDGE + NEDGE + e];
  } else {
    s = d = e - NEDGE;
  }
  const float wgt = alphabuf[e * NHEAD + h] / denom[d * NHEAD + h];
  atomicAdd(&aggr[(size_t)d * NF + f], xl[(size_t)s * NF + f] * wgt);
}

// ---------------------------------------------------------------------------
// K7: per-node: mean over heads of (aggr/deg) + conv bias
// ---------------------------------------------------------------------------
__global__ __launch_bounds__(256) void k_nodeout(const float* __restrict__ aggr,
                                                 const float* __restrict__ cnt,
                                                 const float* __restrict__ cb,
                                                 float* __restrict__ nodeout, int l) {
  const int idx = blockIdx.x * blockDim.x + threadIdx.x;
  if (idx >= NNODE * NCH) return;
  const int n = idx >> 5, c = idx & 31;
  const float deg = cnt[n] + 1.0f;  // + self loop
  const float o = 0.5f * (aggr[(size_t)n * NF + c] + aggr[(size_t)n * NF + NCH + c]) / deg
                + cb[l * NCH + c];
  nodeout[idx] = o;
}

// ---------------------------------------------------------------------------
// K8: BatchNorm batch stats (one block per channel; deterministic tree)
// ---------------------------------------------------------------------------
__global__ __launch_bounds__(256) void k_bnstats(const float* __restrict__ nodeout,
                                                 float* __restrict__ mu,
                                                 float* __restrict__ var) {
  __shared__ float s1[256], s2[256];
  const int c = blockIdx.x;
  const int t = threadIdx.x;
  float a = 0.0f, b = 0.0f;
  for (int n = t; n < NNODE; n += 256) {
    const float v = nodeout[n * NCH + c];
    a += v;
    b += v * v;
  }
  s1[t] = a;
  s2[t] = b;
  __syncthreads();
  for (int st = 128; st > 0; st >>= 1) {
    if (t < st) { s1[t] += s1[t + st]; s2[t] += s2[t + st]; }
    __syncthreads();
  }
  if (t == 0) {
    const float m = s1[0] * (1.0f / NNODE);
    mu[c] = m;
    var[c] = s2[0] * (1.0f / NNODE) - m * m;
  }
}

// ---------------------------------------------------------------------------
// K9: normalize + leaky_relu(0.01) into Hmat[:, l*32 + c]
// ---------------------------------------------------------------------------
__global__ __launch_bounds__(256) void k_norm(const float* __restrict__ nodeout,
                                              const float* __restrict__ mu,
                                              const float* __restrict__ var,
                                              const float* __restrict__ gamma,
                                              const float* __restrict__ beta,
                                              float* __restrict__ Hmat, int l) {
  const int idx = blockIdx.x * blockDim.x + threadIdx.x;
  if (idx >= NNODE * NCH) return;
  const int n = idx >> 5, c = idx & 31;
  float v = gamma[l * NCH + c] * (nodeout[idx] - mu[c]) * rsqrtf(var[c] + EPSBN)
          + beta[l * NCH + c];
  v = v > 0.0f ? v : 0.01f * v;
  Hmat[(size_t)n * DMLP + l * NCH + c] = v;
}

// ---------------------------------------------------------------------------
// K10: fused MLP head with fp32 WMMA (V_WMMA_F32_16X16X4_F32).
//
// Block = 10 waves = one 160-row m-panel; 125 blocks cover all 20000 rows
// exactly (every wave active -> EXEC all-1s, clean barriers). Per 16-col
// W1 tile, the block cooperatively stages the 800x16 B panel into 51.2 KB
// of LDS, packed as K-pairs so each lane's (B[ka][n], B[ka+1][n]) fragment
// is a single aligned ds_load_b64 (ka is always even). Inner loop is then
// 1 global_load_b64 (A, L2-resident) + 1 ds_load_b64 (B) per WMMA.
// Second GEMM (800x5) folded into per-lane partials, reduced with shfl_xor
// inside each 16-lane half. No atomics -> deterministic.
// ---------------------------------------------------------------------------
__global__ __launch_bounds__(MLP_THREADS) void k_mlp(const float* __restrict__ Hm,
                                                     const float* __restrict__ W1,
                                                     const float* __restrict__ b1,
                                                     const float* __restrict__ W2,
                                                     const float* __restrict__ b2,
                                                     float* __restrict__ out) {
  __shared__ __align__(16) float ldsB[(DMLP / 2) * 32];  // 400*32 floats = 51.2 KB

  const int tid = threadIdx.x;
  const int widb = tid >> 5;      // wave in block: 0..9
  const int lane = tid & 31;
  const int half = lane >> 4;     // which 16-lane half
  const int lm = lane & 15;       // row (A) / col (B,C)
  const int m0 = (blockIdx.x * MLP_WAVES + widb) * 16;
  const float* arow = Hm + (size_t)(m0 + lm) * DMLP;

  float p[8][5];
#pragma unroll
  for (int i = 0; i < 8; ++i)
#pragma unroll
    for (int j = 0; j < 5; ++j) p[i][j] = 0.0f;

  for (int nt = 0; nt < DMLP / 16; ++nt) {
    const int n0 = nt * 16;

    __syncthreads();  // previous iteration's readers done before restaging
    // Stage B panel: 12800 elements / 320 threads = 40 each, exact.
#pragma unroll 4
    for (int i = tid; i < DMLP * 16; i += MLP_THREADS) {
      const int k = i >> 4;
      const int n = i & 15;
      ldsB[(k >> 1) * 32 + n * 2 + (k & 1)] = W1[k * DMLP + n0 + n];
    }
    __syncthreads();

    v8f c = {0.f, 0.f, 0.f, 0.f, 0.f, 0.f, 0.f, 0.f};
    for (int k0 = 0; k0 < DMLP; k0 += 4) {
      const int ka = k0 + 2 * half;  // even; A: K=2*half+{0,1}; B mirrors (ISA 7.12.2)
      const v2f a = *(const v2f*)(arow + ka);
      const v2f b = *(const v2f*)(&ldsB[(ka >> 1) * 32 + lm * 2]);
      c = __builtin_amdgcn_wmma_f32_16x16x4_f32(false, a, false, b,
                                                (short)0, c, false, false);
    }

    const float bias = b1[n0 + lm];
    float w2r[5];
#pragma unroll
    for (int j = 0; j < 5; ++j) w2r[j] = W2[(n0 + lm) * 5 + j];
#pragma unroll
    for (int i = 0; i < 8; ++i) {
      float v = c[i] + bias;
      v = v > 0.0f ? v : 0.0f;  // relu
#pragma unroll
      for (int j = 0; j < 5; ++j) p[i][j] += v * w2r[j];
    }
  }

  // reduce over the 16 columns held across each 16-lane half
#pragma unroll
  for (int i = 0; i < 8; ++i)
#pragma unroll
    for (int j = 0; j < 5; ++j) {
      float v = p[i][j];
      v += __shfl_xor(v, 1);
      v += __shfl_xor(v, 2);
      v += __shfl_xor(v, 4);
      v += __shfl_xor(v, 8);
      p[i][j] = v;
    }
  if (lm == 0) {
#pragma unroll
    for (int i = 0; i < 8; ++i) {
      const int m = m0 + i + 8 * half;  // C/D: vgpr i -> row i+8*half
#pragma unroll
      for (int j = 0; j < 5; ++j) out[m * 5 + j] = p[i][j] + b2[j];
    }
  }
}

// ---------------------------------------------------------------------------
// Host launcher
// ---------------------------------------------------------------------------
extern "C" void kernel_launch(void* const* d_in, const int* in_sizes, int n_in,
                              void* d_out, int out_size, void* d_ws, size_t ws_size,
                              hipStream_t stream) {
  const float* x   = (const float*)d_in[0];
  const int*   ei  = (const int*)d_in[1];
  const float* ew  = (const float*)d_in[2];
  const float* Wl  = (const float*)d_in[3];
  const float* bl  = (const float*)d_in[4];
  const float* Wr  = (const float*)d_in[5];
  const float* br  = (const float*)d_in[6];
  const float* We  = (const float*)d_in[7];
  const float* att = (const float*)d_in[8];
  const float* cb  = (const float*)d_in[9];
  const float* g1  = (const float*)d_in[10];
  const float* be1 = (const float*)d_in[11];
  const float* W1  = (const float*)d_in[12];
  const float* b1  = (const float*)d_in[13];
  const float* W2  = (const float*)d_in[14];
  const float* b2  = (const float*)d_in[15];
  float* out = (float*)d_out;

  float* ws = (float*)d_ws;
  size_t o = 0;
  float* xl       = ws + o; o += (size_t)NNODE * NF;
  float* xr       = ws + o; o += (size_t)NNODE * NF;
  float* aggr     = ws + o; o += (size_t)NNODE * NF;
  float* cnt      = ws + o; o += NNODE;
  float* lasum    = ws + o; o += NNODE;
  float* amax     = ws + o; o += NNODE * NHEAD;
  float* denom    = ws + o; o += NNODE * NHEAD;
  float* alphabuf = ws + o; o += (size_t)(NEDGE + NNODE) * NHEAD;
  float* nodeout  = ws + o; o += (size_t)NNODE * NCH;
  float* mu       = ws + o; o += NCH;
  float* var      = ws + o; o += NCH;
  float* Hmat     = ws + o; o += (size_t)NNODE * DMLP;

  const dim3 blk(256);
  const int gNF   = (NNODE * NF + 255) / 256;
  const int gE    = (NEDGE + 255) / 256;
  const int gEH   = ((NEDGE + NNODE) * NHEAD + 255) / 256;
  const int gEF   = ((NEDGE + NNODE) * NF + 255) / 256;
  const int gNC   = (NNODE * NCH + 255) / 256;

  for (int l = 0; l < NLAYER; ++l) {
    k_init<<<gNF, blk, 0, stream>>>(aggr, cnt, lasum, amax, denom);
    k_transform<<<gNF, blk, 0, stream>>>(x, Wl, bl, Wr, br, xl, xr, l);
    k_cnt<<<gE, blk, 0, stream>>>(ei, ew, cnt, lasum, l);
    k_att1<<<gEH, blk, 0, stream>>>(ei, ew, xl, xr, We, att, cnt, lasum,
                                    alphabuf, amax, l);
    k_att2<<<gEH, blk, 0, stream>>>(ei, alphabuf, amax, denom, l);
    k_scatter<<<gEF, blk, 0, stream>>>(ei, xl, alphabuf, denom, aggr, l);
    k_nodeout<<<gNC, blk, 0, stream>>>(aggr, cnt, cb, nodeout, l);
    k_bnstats<<<NCH, blk, 0, stream>>>(nodeout, mu, var);
    k_norm<<<gNC, blk, 0, stream>>>(nodeout, mu, var, g1, be1, Hmat, l);
  }

  // 125 blocks x 10 waves = 1250 m-tiles (16 rows each) = 20000 rows exactly
  k_mlp<<<NNODE / MLP_ROWS, dim3(MLP_THREADS), 0, stream>>>(Hmat, W1, b1, W2, b2, out);
}